// CGCNReg_1563368096537
// MI455X (gfx1250) — compile-verified
//
#include <hip/hip_runtime.h>

// ---------------------------------------------------------------------------
// 2-layer GCN (PyG GCNConv semantics) for MI455X / gfx1250.
//   L1: h = relu( dinv ⊙ scatter_add(dinv[src]·(X@W1)[src] -> dst) + b1 )
//   L2: out = dinv ⊙ scatter_add(dinv[src]·(h@W2)[src] -> dst) + b2
// dinv = rsqrt(in_degree + 1); self-loops folded into accumulator init.
// GEMM X@W1 runs on V_WMMA_F32_16X16X4_F32 (fp32 tensor path, wave32).
// ---------------------------------------------------------------------------

typedef float v2f __attribute__((ext_vector_type(2)));
typedef float v8f __attribute__((ext_vector_type(8)));

#define F_IN 512
#define HDIM 256

// ---- degree / normalization ------------------------------------------------

__global__ void k_init_deg(float* __restrict__ deg, int n) {
    int i = blockIdx.x * blockDim.x + threadIdx.x;
    if (i < n) deg[i] = 1.0f;  // self-loop contribution
}

__global__ void k_deg_accum(const int* __restrict__ dst, float* __restrict__ deg, int e) {
    int i = blockIdx.x * blockDim.x + threadIdx.x;
    if (i < e) atomicAdd(&deg[dst[i]], 1.0f);
}

__global__ void k_dinv(float* __restrict__ deg, int n) {  // in-place deg -> rsqrt(deg)
    int i = blockIdx.x * blockDim.x + threadIdx.x;
    if (i < n) deg[i] = rsqrtf(deg[i]);
}

// ---- Layer-1 GEMM: T1s = dinv ⊙ (X @ W1); acc initialized with self-loop ---
// Each wave: one 16x64 strip (4 WMMA accumulators). Block = 8 waves = 128 rows.

__global__ __launch_bounds__(256)
void k_gemm1_wmma(const float* __restrict__ X, const float* __restrict__ W1,
                  const float* __restrict__ dinv, float* __restrict__ T1s,
                  float* __restrict__ acc, int n) {
    const int wave  = threadIdx.x >> 5;
    const int lane  = threadIdx.x & 31;
    const int row16 = lane & 15;
    const int khalf = lane >> 4;            // 0: K={0,1}, 1: K={2,3}

    const int m0 = (blockIdx.x * 8 + wave) * 16;
    const int n0 = blockIdx.y * 64;
    if (m0 >= n) return;                    // wave-uniform exit (tail block)

    int arow = m0 + row16;
    if (arow > n - 1) arow = n - 1;         // safety clamp (N%16==0 here)
    const float* aptr = X + (size_t)arow * F_IN + khalf * 2;

    v8f c0 = {}, c1 = {}, c2 = {}, c3 = {};

#pragma unroll 4
    for (int k = 0; k < F_IN; k += 4) {
        // A tile 16x4: v0 = K (0|2), v1 = K (1|3) per half-wave -> one b64 load
        v2f a = *(const v2f*)(aptr + k);

        // B tile 4x16 per n-subtile: lanes 0-15 carry cols, halves carry K
        const float* bbase = W1 + (size_t)(k + 2 * khalf) * HDIM + n0 + row16;
        v2f b0; b0.x = bbase[0];   b0.y = bbase[HDIM + 0];
        v2f b1; b1.x = bbase[16];  b1.y = bbase[HDIM + 16];
        v2f b2; b2.x = bbase[32];  b2.y = bbase[HDIM + 32];
        v2f b3; b3.x = bbase[48];  b3.y = bbase[HDIM + 48];

        c0 = __builtin_amdgcn_wmma_f32_16x16x4_f32(false, a, false, b0, (short)0, c0, false, false);
        c1 = __builtin_amdgcn_wmma_f32_16x16x4_f32(false, a, false, b1, (short)0, c1, false, false);
        c2 = __builtin_amdgcn_wmma_f32_16x16x4_f32(false, a, false, b2, (short)0, c2, false, false);
        c3 = __builtin_amdgcn_wmma_f32_16x16x4_f32(false, a, false, b3, (short)0, c3, false, false);
    }

    // Epilogue: scale row by dinv[row]; write message table AND accumulator
    // (acc init == self-loop message dinv[i]*(XW1)[i]).
#pragma unroll
    for (int j = 0; j < 8; ++j) {
        int row = m0 + j + 8 * khalf;       // C/D layout: vgpr j -> M=j / M=j+8
        if (row < n) {
            float s = dinv[row];
            size_t base = (size_t)row * HDIM + n0 + row16;
            float v;
            v = c0[j] * s; T1s[base +  0] = v; acc[base +  0] = v;
            v = c1[j] * s; T1s[base + 16] = v; acc[base + 16] = v;
            v = c2[j] * s; T1s[base + 32] = v; acc[base + 32] = v;
            v = c3[j] * s; T1s[base + 48] = v; acc[base + 48] = v;
        }
    }
}

// ---- Layer-1 edge scatter: acc[dst] += T1s[src]  (64 threads / edge) -------

__global__ __launch_bounds__(256)
void k_scatter1(const int* __restrict__ src, const int* __restrict__ dst,
                const float* __restrict__ T1s, float* __restrict__ acc, int e) {
    int gid  = blockIdx.x * 256 + threadIdx.x;
    int eidx = gid >> 6;
    if (eidx >= e) return;
    int f = (gid & 63) * 4;
    int s = src[eidx];
    int d = dst[eidx];
    float4 m = *(const float4*)(T1s + (size_t)s * HDIM + f);
    float* ab = acc + (size_t)d * HDIM + f;
    atomicAdd(ab + 0, m.x);
    atomicAdd(ab + 1, m.y);
    atomicAdd(ab + 2, m.z);
    atomicAdd(ab + 3, m.w);
}

// ---- Layer 2: fused relu(dinv*acc + b1) · W2, pre-scaled by dinv -----------
// Wave per node; 8 features per lane; cross-lane reduce (wave32).

__global__ __launch_bounds__(256)
void k_layer2_dot(const float* __restrict__ acc, const float* __restrict__ b1,
                  const float* __restrict__ W2, const float* __restrict__ dinv,
                  float* __restrict__ t2s, float* __restrict__ acc2, int n) {
    int node = blockIdx.x * 8 + (threadIdx.x >> 5);
    int lane = threadIdx.x & 31;
    if (node >= n) return;

    float di = dinv[node];
    int f = lane * 8;
    const float4* ap = (const float4*)(acc + (size_t)node * HDIM + f);
    const float4* bp = (const float4*)(b1 + f);
    const float4* wp = (const float4*)(W2 + f);
    float4 a0 = ap[0], a1 = ap[1];
    float4 bb0 = bp[0], bb1 = bp[1];
    float4 w0 = wp[0], w1 = wp[1];

    float s = fmaxf(di * a0.x + bb0.x, 0.f) * w0.x
            + fmaxf(di * a0.y + bb0.y, 0.f) * w0.y
            + fmaxf(di * a0.z + bb0.z, 0.f) * w0.z
            + fmaxf(di * a0.w + bb0.w, 0.f) * w0.w
            + fmaxf(di * a1.x + bb1.x, 0.f) * w1.x
            + fmaxf(di * a1.y + bb1.y, 0.f) * w1.y
            + fmaxf(di * a1.z + bb1.z, 0.f) * w1.z
            + fmaxf(di * a1.w + bb1.w, 0.f) * w1.w;

#pragma unroll
    for (int off = 16; off >= 1; off >>= 1)
        s += __shfl_xor(s, off, 32);

    if (lane == 0) {
        float t = di * s;          // pre-scale by dinv[src]
        t2s[node]  = t;
        acc2[node] = t;            // self-loop init
    }
}

// ---- Layer-2 edge scatter (scalar) -----------------------------------------

__global__ void k_scatter2(const int* __restrict__ src, const int* __restrict__ dst,
                           const float* __restrict__ t2s, float* __restrict__ acc2, int e) {
    int i = blockIdx.x * blockDim.x + threadIdx.x;
    if (i < e) atomicAdd(&acc2[dst[i]], t2s[src[i]]);
}

// ---- Final: out = dinv * acc2 + b2 -----------------------------------------

__global__ void k_final(const float* __restrict__ acc2, const float* __restrict__ dinv,
                        const float* __restrict__ b2, float* __restrict__ out, int n) {
    int i = blockIdx.x * blockDim.x + threadIdx.x;
    if (i < n) out[i] = dinv[i] * acc2[i] + b2[0];
}

// ---------------------------------------------------------------------------

extern "C" void kernel_launch(void* const* d_in, const int* in_sizes, int n_in,
                              void* d_out, int out_size, void* d_ws, size_t ws_size,
                              hipStream_t stream) {
    const float* X  = (const float*)d_in[0];
    const int*   ei = (const int*)d_in[1];
    const float* W1 = (const float*)d_in[2];
    const float* b1 = (const float*)d_in[3];
    const float* W2 = (const float*)d_in[4];
    const float* b2 = (const float*)d_in[5];
    float* out = (float*)d_out;

    const int n = in_sizes[0] / F_IN;   // 100000
    const int e = in_sizes[1] / 2;      // 3200000
    const int* src = ei;                // edge_index row 0
    const int* dst = ei + e;            // edge_index row 1

    // workspace layout (floats), 16B-aligned slices
    float* ws   = (float*)d_ws;
    size_t feat = (size_t)n * HDIM;
    float* T1s  = ws;                   // [N,256] dinv-scaled XW1
    float* acc  = ws + feat;            // [N,256] layer-1 accumulator
    float* dinv = ws + 2 * feat;        // [N] deg -> rsqrt in place
    float* t2s  = dinv + n;             // [N]
    float* acc2 = t2s + n;              // [N]

    const int tpb = 256;

    k_init_deg <<<(n + tpb - 1) / tpb, tpb, 0, stream>>>(dinv, n);
    k_deg_accum<<<(e + tpb - 1) / tpb, tpb, 0, stream>>>(dst, dinv, e);
    k_dinv     <<<(n + tpb - 1) / tpb, tpb, 0, stream>>>(dinv, n);

    int mtiles = (n + 15) / 16;                  // 6250
    dim3 grid1((mtiles + 7) / 8, HDIM / 64);     // 782 x 4
    k_gemm1_wmma<<<grid1, 256, 0, stream>>>(X, W1, dinv, T1s, acc, n);

    long long work1 = (long long)e * 64;         // 64 threads per edge
    k_scatter1<<<(unsigned)((work1 + 255) / 256), 256, 0, stream>>>(src, dst, T1s, acc, e);

    k_layer2_dot<<<(n + 7) / 8, 256, 0, stream>>>(acc, b1, W2, dinv, t2s, acc2, n);
    k_scatter2<<<(e + tpb - 1) / tpb, tpb, 0, stream>>>(src, dst, t2s, acc2, e);
    k_final   <<<(n + tpb - 1) / tpb, tpb, 0, stream>>>(acc2, dinv, b2, out, n);
}